// AdaptiveGraphStructure_25099788878456
// MI455X (gfx1250) — compile-verified
//
#include <hip/hip_runtime.h>
#include <hip/hip_bf16.h>
#include <cstdint>

// Problem constants (reference: B=2, N=1024, D=128, H=D/2=64)
#define B_ 2
#define N_ 1024
#define D_ 128
#define H_ 64

typedef __attribute__((ext_vector_type(2))) float v2f;
typedef __attribute__((ext_vector_type(8))) float v8f;

// ---------------------------------------------------------------------------
// Kernel 1: projections  ai = X @ Wi^T , aj = X @ Wj^T  (full f32 WMMA)
// One wave (32 threads) per 16-row tile of X. Uses V_WMMA_F32_16X16X4_F32:
//   A (16x4 f32): lane L holds row M=L&15; v0/v1 hold K = 2*(L>>4) .. +1
//   B (4x16 f32): lane L holds col N=L&15; v0/v1 hold K = 2*(L>>4) .. +1
//   C (16x16 f32): VGPR r -> row M = r + 8*(L>>4), col N = L&15
// ---------------------------------------------------------------------------
__global__ __launch_bounds__(32) void proj_wmma(
    const float* __restrict__ X,    // (B, N, D)
    const float* __restrict__ W1,   // (H, 2*D)  [Wi | Wj]
    float* __restrict__ ai,         // (B, N, H)
    float* __restrict__ aj)         // (B, N, H)
{
  const int tile = blockIdx.x;              // 0 .. B*N/16 - 1
  const int b    = tile / (N_ / 16);
  const int i0   = (tile % (N_ / 16)) * 16;
  const int lane = threadIdx.x;             // 0..31 (wave32)
  const int m    = lane & 15;
  const int kh   = lane >> 4;               // selects K pair {0,1} vs {2,3}

  const float* Xrow = X + ((size_t)(b * N_ + i0 + m)) * D_;

  for (int sel = 0; sel < 2; ++sel) {       // 0 -> ai (Wi), 1 -> aj (Wj)
    float* dst = sel ? aj : ai;
    for (int c = 0; c < 4; ++c) {           // 4 column tiles of 16 -> H=64
      const int h = c * 16 + m;             // output feature (B-matrix col)
      const float* Wrow = W1 + (size_t)h * (2 * D_) + sel * D_;
      v8f acc = {};
      #pragma unroll 4
      for (int kk = 0; kk < D_; kk += 4) {  // 32 K-steps of 4
        v2f a, bm;
        a.x  = Xrow[kk + 2 * kh];
        a.y  = Xrow[kk + 2 * kh + 1];
        bm.x = Wrow[kk + 2 * kh];
        bm.y = Wrow[kk + 2 * kh + 1];
        acc = __builtin_amdgcn_wmma_f32_16x16x4_f32(
            false, a, false, bm, (short)0, acc, false, false);
      }
      #pragma unroll
      for (int r = 0; r < 8; ++r) {
        dst[((size_t)(b * N_ + i0 + r + 8 * kh)) * H_ + c * 16 + m] = acc[r];
      }
    }
  }
}

// ---------------------------------------------------------------------------
// Kernel 2: fused pairwise MLP score + mask + row softmax.
// One block = 8 output rows (same batch). Each thread keeps its aj row in
// registers (16 float4) and reuses it across the 8 i-rows -> 8x less L2
// traffic. Softmax per row is done by one wave32 with shuffles.
// ---------------------------------------------------------------------------
__global__ __launch_bounds__(256) void edge_softmax(
    const float* __restrict__ ai,        // (B, N, H)
    const float* __restrict__ aj,        // (B, N, H)
    const unsigned char* __restrict__ visited,  // (B, N) bool (1 byte)
    const float* __restrict__ b1,        // (H,)
    const float* __restrict__ W2,        // (1, H)
    const float* __restrict__ b2,        // (1,)
    float* __restrict__ out)             // (B, N, N)
{
  __shared__ float4 s_ai[8][H_ / 4];     // ai rows + b1
  __shared__ float4 s_w2[H_ / 4];
  __shared__ float  s_ew[8][N_];         // edge weights (32 KB)
  __shared__ int    s_vis[8];
  __shared__ float  s_b2;

  const int blocksPerBatch = N_ / 8;
  const int b  = blockIdx.x / blocksPerBatch;
  const int i0 = (blockIdx.x % blocksPerBatch) * 8;
  const int t  = threadIdx.x;

  for (int idx = t; idx < 8 * H_; idx += 256) {
    const int il = idx / H_, h = idx % H_;
    ((float*)s_ai)[idx] = ai[((size_t)(b * N_ + i0 + il)) * H_ + h] + b1[h];
  }
  if (t < H_) ((float*)s_w2)[t] = W2[t];
  if (t < 8)  s_vis[t] = visited[b * N_ + i0 + t] ? 1 : 0;
  if (t == 0) s_b2 = b2[0];
  __syncthreads();

  const float b2v = s_b2;
  for (int j = t; j < N_; j += 256) {
    const float4* ajr = (const float4*)(aj + ((size_t)(b * N_ + j)) * H_);
    float4 ar[H_ / 4];
    #pragma unroll
    for (int q = 0; q < H_ / 4; ++q) ar[q] = ajr[q];
    const int vj = visited[b * N_ + j] ? 1 : 0;
    #pragma unroll
    for (int il = 0; il < 8; ++il) {
      float s;
      if (s_vis[il] | vj) {
        s = -1.0e9f;
      } else {
        s = b2v;
        #pragma unroll
        for (int q = 0; q < H_ / 4; ++q) {
          const float4 av = s_ai[il][q];
          const float4 wv = s_w2[q];
          s += wv.x * fmaxf(av.x + ar[q].x, 0.0f);
          s += wv.y * fmaxf(av.y + ar[q].y, 0.0f);
          s += wv.z * fmaxf(av.z + ar[q].z, 0.0f);
          s += wv.w * fmaxf(av.w + ar[q].w, 0.0f);
        }
      }
      s_ew[il][j] = s;
    }
  }
  __syncthreads();

  // One wave32 per row: max, exp-sum, normalize (warpSize == 32 on gfx1250).
  const int r    = t >> 5;
  const int lane = t & 31;
  float m = -3.4e38f;
  #pragma unroll 4
  for (int k = 0; k < N_ / 32; ++k) m = fmaxf(m, s_ew[r][lane + 32 * k]);
  for (int off = 16; off > 0; off >>= 1) m = fmaxf(m, __shfl_xor(m, off, 32));

  float sum = 0.0f;
  #pragma unroll 4
  for (int k = 0; k < N_ / 32; ++k) {
    const int j = lane + 32 * k;
    const float e = __expf(s_ew[r][j] - m);
    s_ew[r][j] = e;
    sum += e;
  }
  for (int off = 16; off > 0; off >>= 1) sum += __shfl_xor(sum, off, 32);
  const float inv = 1.0f / sum;

  float* orow = out + ((size_t)(b * N_ + i0 + r)) * N_;
  #pragma unroll 4
  for (int k = 0; k < N_ / 32; ++k) {
    const int j = lane + 32 * k;
    orow[j] = s_ew[r][j] * inv;   // coalesced: lane-consecutive j
  }
}

extern "C" void kernel_launch(void* const* d_in, const int* in_sizes, int n_in,
                              void* d_out, int out_size, void* d_ws, size_t ws_size,
                              hipStream_t stream) {
  // Inputs (setup_inputs order):
  // 0: node_embeddings (B,N,D) f32   1: visited (B,N) bool(1B)
  // 2: remaining_capacity (unused)   3: W1 (H,2D) f32
  // 4: b1 (H,) f32  5: W2 (1,H) f32  6: b2 (1,) f32
  const float*         node    = (const float*)d_in[0];
  const unsigned char* visited = (const unsigned char*)d_in[1];
  const float*         W1      = (const float*)d_in[3];
  const float*         b1      = (const float*)d_in[4];
  const float*         W2      = (const float*)d_in[5];
  const float*         b2      = (const float*)d_in[6];
  float*               out     = (float*)d_out;

  // Workspace: ai + aj = 2 * B*N*H * 4B = 1 MB
  float* ai = (float*)d_ws;
  float* aj = ai + (size_t)B_ * N_ * H_;

  proj_wmma<<<B_ * (N_ / 16), 32, 0, stream>>>(node, W1, ai, aj);
  edge_softmax<<<B_ * (N_ / 8), 256, 0, stream>>>(ai, aj, visited, b1, W2, b2, out);
}